// BackgroundLoss_82386062672210
// MI455X (gfx1250) — compile-verified
//
#include <hip/hip_runtime.h>

typedef __attribute__((ext_vector_type(2))) float v2f;
typedef __attribute__((ext_vector_type(8))) float v8f;

// Workspace layout (d_ws):
//   [0..63]   bytes : float acc[16]  -- acc[0]=noise_sum, acc[1]=nb, acc[2]=attr_sum, acc[3]=n_valid
//   [64.. )   bytes : u32 maxbits[roundup(num_pids,64)]
//                     0 = pid absent, else as_uint(max_beta)+1  (monotonic for beta>=0)

__global__ void bg_init_82386062672210(unsigned int* __restrict__ maxbits,
                                       float* __restrict__ acc,
                                       const int* __restrict__ num_pids_p) {
    const int P = *num_pids_p;
    const int Pround = (P + 63) & ~63;          // zero padded tail too -> finalize loads are unconditional
    const int tid = blockIdx.x * blockDim.x + threadIdx.x;
    if (blockIdx.x == 0 && threadIdx.x < 16) acc[threadIdx.x] = 0.0f;
    for (int p = tid; p < Pround; p += gridDim.x * blockDim.x) maxbits[p] = 0u;
}

__device__ __forceinline__ void bg_hit(float b, int p, int P,
                                       unsigned int* __restrict__ maxbits,
                                       float* __restrict__ acc) {
    if (p == 0) {
        atomicAdd(&acc[0], b);
        atomicAdd(&acc[1], 1.0f);
    } else if ((unsigned)p < (unsigned)P) {
        atomicMax(&maxbits[p], __float_as_uint(b) + 1u);
    }
}

__global__ void bg_scatter_82386062672210(const float* __restrict__ beta,
                                          const int* __restrict__ pid,
                                          int n,
                                          unsigned int* __restrict__ maxbits,
                                          float* __restrict__ acc,
                                          const int* __restrict__ num_pids_p) {
    const int P = *num_pids_p;
    const int tid = blockIdx.x * blockDim.x + threadIdx.x;
    const int nthreads = gridDim.x * blockDim.x;
    const int n4 = n >> 2;
    const float4* __restrict__ b4 = (const float4*)beta;
    const int4*   __restrict__ p4 = (const int4*)pid;
    for (int j = tid; j < n4; j += nthreads) {           // global_load_b128 streaming
        float4 b = b4[j];
        int4   p = p4[j];
        bg_hit(b.x, p.x, P, maxbits, acc);
        bg_hit(b.y, p.y, P, maxbits, acc);
        bg_hit(b.z, p.z, P, maxbits, acc);
        bg_hit(b.w, p.w, P, maxbits, acc);
    }
    for (int i = (n4 << 2) + tid; i < n; i += nthreads)  // tail
        bg_hit(beta[i], pid[i], P, maxbits, acc);
}

// Finalize over P segments through the matrix pipe: D = A x ones(4x16) + C.
// Each V_WMMA_F32_16X16X4_F32 sums 64 f32 values exactly (ones-B makes any
// element->A-slot bijection valid). Table is zero-padded to 64, so loads are
// unconditional b64; only predicate left is p!=0 (noise segment).
__global__ void bg_finalize_82386062672210(const unsigned int* __restrict__ maxbits,
                                           const int* __restrict__ num_pids_p,
                                           float* __restrict__ acc) {
    __shared__ float s_sum[8];
    __shared__ float s_cnt[8];
    const int P    = *num_pids_p;
    const int lane = threadIdx.x & 31;
    const int wave = threadIdx.x >> 5;
    const int wavesPerBlock = blockDim.x >> 5;
    const int globalWave = blockIdx.x * wavesPerBlock + wave;
    const int numWaves   = gridDim.x * wavesPerBlock;
    const int numTiles   = (P + 63) >> 6;

    const uint2* __restrict__ mb2 = (const uint2*)maxbits;

    v8f c_sum = {};
    v8f c_cnt = {};
    v2f ones;
    ones.x = 1.0f;
    ones.y = 1.0f;

    for (int t = globalWave; t < numTiles; t += numWaves) {  // wave-uniform trip count
        const int base = t * 64 + lane * 2;
        const uint2 m = mb2[t * 32 + lane];                  // global_load_b64, always in range

        const bool ok0 = (m.x != 0u) && (base != 0);         // base==0 is the pid-0 slot
        const bool ok1 = (m.y != 0u);                        // base+1 is never 0
        v2f av, ac;
        av.x = ok0 ? 1.0f - __uint_as_float(m.x - 1u) : 0.0f;
        av.y = ok1 ? 1.0f - __uint_as_float(m.y - 1u) : 0.0f;
        ac.x = ok0 ? 1.0f : 0.0f;
        ac.y = ok1 ? 1.0f : 0.0f;

        // EXEC is all-ones here; 64 values accumulated per op.
        c_sum = __builtin_amdgcn_wmma_f32_16x16x4_f32(false, av, false, ones,
                                                      (short)0, c_sum, false, false);
        c_cnt = __builtin_amdgcn_wmma_f32_16x16x4_f32(false, ac, false, ones,
                                                      (short)0, c_cnt, false, false);
    }

    // Every D column of a row is identical (B==1). Lanes 0..15 hold rows 0..7,
    // lanes 16..31 hold rows 8..15.  total = lane0 + lane16 partials.
    float ps = 0.0f, pc = 0.0f;
    #pragma unroll
    for (int r = 0; r < 8; ++r) { ps += c_sum[r]; pc += c_cnt[r]; }
    const float wave_sum = __shfl(ps, 0, 32) + __shfl(ps, 16, 32);
    const float wave_cnt = __shfl(pc, 0, 32) + __shfl(pc, 16, 32);

    if (lane == 0) { s_sum[wave] = wave_sum; s_cnt[wave] = wave_cnt; }
    __syncthreads();
    if (threadIdx.x == 0) {
        float bs = 0.0f, bc = 0.0f;
        for (int w = 0; w < wavesPerBlock; ++w) { bs += s_sum[w]; bc += s_cnt[w]; }
        if (bs != 0.0f || bc != 0.0f) {
            atomicAdd(&acc[2], bs);
            atomicAdd(&acc[3], bc);
        }
    }
}

__global__ void bg_final_82386062672210(const float* __restrict__ acc,
                                        float* __restrict__ out) {
    if (blockIdx.x == 0 && threadIdx.x == 0) {
        const float noise_sum = acc[0];
        const float nb        = acc[1];
        const float attr_sum  = acc[2];
        const float n_valid   = acc[3];
        const float attractive = attr_sum / fmaxf(n_valid, 1.0f);
        const float noise      = 0.1f * noise_sum / fmaxf(nb, 1.0f);
        out[0] = (nb == 0.0f) ? 0.0f : (attractive + noise);
    }
}

extern "C" void kernel_launch(void* const* d_in, const int* in_sizes, int n_in,
                              void* d_out, int out_size, void* d_ws, size_t ws_size,
                              hipStream_t stream) {
    // inputs: 0=w, 1=beta, 2=x, 3=y, 4=particle_id, 5=num_pids (device scalar)
    const float* beta      = (const float*)d_in[1];
    const int*   pid       = (const int*)d_in[4];
    const int*   num_pids  = (const int*)d_in[5];
    const int    n         = in_sizes[1];

    float*        acc     = (float*)d_ws;
    unsigned int* maxbits = (unsigned int*)((char*)d_ws + 64);
    float*        out     = (float*)d_out;

    bg_init_82386062672210<<<1024, 256, 0, stream>>>(maxbits, acc, num_pids);

    int blocks = (n / 4 + 255) / 256;           // ~7813 blocks for N=8M (float4 lanes)
    if (blocks < 1) blocks = 1;
    bg_scatter_82386062672210<<<blocks, 256, 0, stream>>>(beta, pid, n, maxbits, acc, num_pids);

    bg_finalize_82386062672210<<<256, 256, 0, stream>>>(maxbits, num_pids, acc);

    bg_final_82386062672210<<<1, 32, 0, stream>>>(acc, out);
}